// CRF_6734508720491
// MI455X (gfx1250) — compile-verified
//
#include <hip/hip_runtime.h>
#include <hip/hip_bf16.h>

#define NTAG   128
#define SEQL   256
#define NBATCH 512
#define TP2    130      // (NTAG + 2) row stride of transitions
#define PADIDX (-1)

typedef __attribute__((ext_vector_type(2))) float v2f;
typedef __attribute__((ext_vector_type(8))) float v8f;

// ---------------------------------------------------------------------------
// Kernel 1: E[i][j] = exp(transitions[i][j]) for i,j < NTAG (constant B matrix
// of the per-step GEMM).
// ---------------------------------------------------------------------------
__global__ void crf_expE(const float* __restrict__ trans, float* __restrict__ E) {
    const int i = blockIdx.x;     // 128 blocks
    const int j = threadIdx.x;    // 128 threads
    E[i * NTAG + j] = __expf(trans[i * TP2 + j]);
}

// ---------------------------------------------------------------------------
// Kernel 2: numerator log-likelihood per batch element (path score).
// ---------------------------------------------------------------------------
__global__ void crf_numer(const float* __restrict__ em,
                          const float* __restrict__ trans,
                          const int*   __restrict__ tags,
                          float* __restrict__ llh) {
    const int b = blockIdx.x * blockDim.x + threadIdx.x;
    if (b >= NBATCH) return;

    const int t0 = tags[b];
    float acc = trans[NTAG * TP2 + max(t0, 0)];       // start -> tags[0]
    int lastTag = max(t0, 0);

    for (int t = 0; t < SEQL; ++t) {
        const int tg = tags[t * NBATCH + b];
        const float m = (tg != PADIDX) ? 1.0f : 0.0f;
        if (tg != PADIDX) lastTag = tg;
        if (t < SEQL - 1)
            acc += em[((size_t)t * NBATCH + b) * NTAG + max(tg, 0)] * m;
        if (t >= 1) {
            const int pg = tags[(t - 1) * NBATCH + b];
            acc += trans[max(pg, 0) * TP2 + max(tg, 0)] * m;
        }
    }
    acc += trans[lastTag * TP2 + (NTAG + 1)];         // last -> end
    const int tgS = tags[(SEQL - 1) * NBATCH + b];
    const float mS = (tgS != PADIDX) ? 1.0f : 0.0f;
    acc += em[((size_t)(SEQL - 1) * NBATCH + b) * NTAG + max(tgS, 0)] * mS;

    llh[b] = acc;
}

// ---------------------------------------------------------------------------
// Kernel 3: forward scan. One WG per 16 batch rows (32 WGs), 8 wave32s.
// Per step t: lp'[b,j] = m_b + log( sum_i exp(lp[b,i]-m_b) * E[i,j] ) + e[t,b,j]
// The sum over i is a (16x128)@(128x128) f32 GEMM done with
// v_wmma_f32_16x16x4_f32; each wave owns one 16-wide column block and keeps
// its B fragments (columns of E) resident in VGPRs for the whole scan.
// Two independent accumulator chains halve the WMMA dependency latency.
// ---------------------------------------------------------------------------
#define LROW 132   // LDS row stride (dwords); 132 % 64 = 4 -> conflict-free column reads

__global__ __launch_bounds__(256, 1)
void crf_forward(const float* __restrict__ em,
                 const float* __restrict__ trans,
                 const int*   __restrict__ tags,
                 const float* __restrict__ E,
                 const float* __restrict__ llh,
                 float* __restrict__ diff) {
    __shared__ float lp[16][LROW];   // log-prob state, 16 batch rows x 128 tags
    __shared__ float P [16][LROW];   // exp(lp - rowmax)
    __shared__ float pm[16][16];     // partial row reductions
    __shared__ float mrow[16];       // per-row max

    const int tid  = threadIdx.x;
    const int lane = tid & 31;
    const int wave = tid >> 5;            // 0..7 -> column block
    const int h    = lane >> 4;           // lane half (A/B K select, D M select)
    const int nl   = lane & 15;           // A: M row, B/D: N col within tile
    const int jcol = wave * 16 + nl;      // global output tag column, 0..127

    const int b0   = blockIdx.x * 16;     // first batch row of this WG
    const int rrow = tid >> 4;            // cooperative phase: row 0..15
    const int c0   = (tid & 15) * 8;      // cooperative phase: 8-col chunk

    // ---- Preload B fragments: bf[kk] holds E[K = kk*4 + 2h + {0,1}][jcol].
    // Constant for the entire 255-step scan (64 VGPRs).
    v2f bf[32];
#pragma unroll
    for (int kk = 0; kk < 32; ++kk) {
        const int kb = kk * 4 + 2 * h;
        bf[kk].x = E[(kb + 0) * NTAG + jcol];
        bf[kk].y = E[(kb + 1) * NTAG + jcol];
    }

    // ---- Init: lp0[b,j] = trans[start, j] + emissions[0, b, j]
    {
        const int b = b0 + rrow;
#pragma unroll
        for (int c = 0; c < 8; ++c) {
            const int j = c0 + c;
            lp[rrow][j] = trans[NTAG * TP2 + j] + em[(size_t)b * NTAG + j];
        }
    }
    __syncthreads();

    // ---- Sequential scan over time (3 barriers per step)
    for (int t = 1; t < SEQL; ++t) {
        // Phase A: per-thread partial row max over an 8-col chunk
        float mx = lp[rrow][c0];
#pragma unroll
        for (int c = 1; c < 8; ++c) mx = fmaxf(mx, lp[rrow][c0 + c]);
        pm[rrow][tid & 15] = mx;
        __syncthreads();

        // Phase B: every thread redundantly reduces its row's 16 partials
        // (cheaper than an extra barrier), then P = exp(lp - rowmax).
        {
            float mr = pm[rrow][0];
#pragma unroll
            for (int i = 1; i < 16; ++i) mr = fmaxf(mr, pm[rrow][i]);
            if ((tid & 15) == 0) mrow[rrow] = mr;   // for the writeback phase
#pragma unroll
            for (int c = 0; c < 8; ++c) {
                const int j = c0 + c;
                P[rrow][j] = __expf(lp[rrow][j] - mr);
            }
            // Warm next step's 8KB emission slab while this step computes.
            if (t + 1 < SEQL)
                __builtin_prefetch(&em[((size_t)(t + 1) * NBATCH + b0 + rrow) * NTAG + c0], 0, 3);
        }
        __syncthreads();

        // Hoist this step's global loads so HBM latency overlaps the WMMAs.
        float ev[8];
        int   tg[8];
#pragma unroll
        for (int r = 0; r < 8; ++r) {
            const int mi = r + 8 * h;
            const int b  = b0 + mi;
            tg[r] = tags[t * NBATCH + b];
            ev[r] = em[((size_t)t * NBATCH + b) * NTAG + jcol];
        }

        // Phase C: 16x16 output tile, K=128 via 2 x 16 chained f32 WMMAs.
        // A fragment (ISA 7.12.2): lane holds (M = nl, K = kb + {0,1}).
        v8f acc0 = {};
        v8f acc1 = {};
#pragma unroll
        for (int kk = 0; kk < 32; kk += 2) {
            const int kb0 = kk * 4 + 2 * h;
            v2f af0 = *(const v2f*)&P[nl][kb0];        // 8B-aligned
            acc0 = __builtin_amdgcn_wmma_f32_16x16x4_f32(
                false, af0, false, bf[kk], (short)0, acc0, false, false);
            const int kb1 = (kk + 1) * 4 + 2 * h;
            v2f af1 = *(const v2f*)&P[nl][kb1];
            acc1 = __builtin_amdgcn_wmma_f32_16x16x4_f32(
                false, af1, false, bf[kk + 1], (short)0, acc1, false, false);
        }

        // Writeback: D lane holds (M = r + 8h, N = nl). Apply log, rowmax,
        // emission, and pad mask.
#pragma unroll
        for (int r = 0; r < 8; ++r) {
            const int mi = r + 8 * h;
            const float newv = mrow[mi] + __logf(acc0[r] + acc1[r]) + ev[r];
            const float oldv = lp[mi][jcol];
            lp[mi][jcol] = (tg[r] != PADIDX) ? newv : oldv;
        }
        __syncthreads();
    }

    // ---- denom[b] = logsumexp_j( lp[b,j] + trans[j, end] ); diff = llh - denom
    {
        float mx = -3.0e38f;
#pragma unroll
        for (int c = 0; c < 8; ++c) {
            const int j = c0 + c;
            mx = fmaxf(mx, lp[rrow][j] + trans[j * TP2 + (NTAG + 1)]);
        }
        pm[rrow][tid & 15] = mx;
        __syncthreads();
        if (tid < 16) {
            float m2 = pm[tid][0];
#pragma unroll
            for (int i = 1; i < 16; ++i) m2 = fmaxf(m2, pm[tid][i]);
            mrow[tid] = m2;
        }
        __syncthreads();
        float s = 0.0f;
        const float mr = mrow[rrow];
#pragma unroll
        for (int c = 0; c < 8; ++c) {
            const int j = c0 + c;
            s += __expf(lp[rrow][j] + trans[j * TP2 + (NTAG + 1)] - mr);
        }
        pm[rrow][tid & 15] = s;
        __syncthreads();
        if (tid < 16) {
            float ss = 0.0f;
#pragma unroll
            for (int i = 0; i < 16; ++i) ss += pm[tid][i];
            const float denom = mrow[tid] + __logf(ss);
            diff[b0 + tid] = llh[b0 + tid] - denom;
        }
    }
}

// ---------------------------------------------------------------------------
// Kernel 4: deterministic mean over the 512 per-batch diffs.
// ---------------------------------------------------------------------------
__global__ void crf_reduce(const float* __restrict__ diff, float* __restrict__ out) {
    __shared__ float s[256];
    const int tid = threadIdx.x;
    s[tid] = diff[tid] + diff[tid + 256];
    __syncthreads();
    for (int st = 128; st > 0; st >>= 1) {
        if (tid < st) s[tid] += s[tid + st];
        __syncthreads();
    }
    if (tid == 0) out[0] = s[0] * (1.0f / (float)NBATCH);
}

// ---------------------------------------------------------------------------
extern "C" void kernel_launch(void* const* d_in, const int* in_sizes, int n_in,
                              void* d_out, int out_size, void* d_ws, size_t ws_size,
                              hipStream_t stream) {
    const float* em    = (const float*)d_in[0];   // (SEQL, NBATCH, NTAG) f32
    const float* trans = (const float*)d_in[1];   // (130, 130) f32
    const int*   tags  = (const int*)  d_in[2];   // (SEQL, NBATCH) i32
    float* out = (float*)d_out;

    // workspace layout
    char* ws = (char*)d_ws;
    float* E    = (float*)(ws);                       // 128*128*4 = 65536 B
    float* llh  = (float*)(ws + 65536);               // 512*4     =  2048 B
    float* diff = (float*)(ws + 65536 + 2048);        // 512*4     =  2048 B

    crf_expE   <<<NTAG, NTAG, 0, stream>>>(trans, E);
    crf_numer  <<<NBATCH / 256, 256, 0, stream>>>(em, trans, tags, llh);
    crf_forward<<<NBATCH / 16, 256, 0, stream>>>(em, trans, tags, E, llh, diff);
    crf_reduce <<<1, 256, 0, stream>>>(diff, out);
}